// VoxelMorphMICCAI2019_22600117912018
// MI455X (gfx1250) — compile-verified
//
#include <hip/hip_runtime.h>

// VoxelMorph 3D trilinear warp, border padding, align_corners=true.
// img: [1,1,160,192,224] f32; phi: [1,3,160,192,224] f32; out: same as img.
//
// Gather-bound: image (27.5 MB) is kept L2-resident; all streamed traffic
// (grid in, output out) is non-temporal. Grid is staged per-wave into LDS via
// the gfx1250 async engine (global_load_async_to_lds_b128 + s_wait_asynccnt),
// gathers use b64 z-pair loads (half the VMEM ops of scalar gathers).

#define D_ 160
#define H_ 192
#define W_ 224
#define N_ (D_ * W_ * H_)
#define HW_ (H_ * W_)

typedef float f32x4 __attribute__((ext_vector_type(4)));
typedef float f32x2 __attribute__((ext_vector_type(2)));

// align-4-safe 8-byte load; backend emits global_load_b64 (unaligned mode) or
// splits into two b32 loads — correct either way.
__device__ __forceinline__ f32x2 load_pair(const float* p) {
    f32x2 r;
    __builtin_memcpy(&r, p, 8);
    return r;
}

__global__ __launch_bounds__(256) void voxelmorph_warp3d_kernel(
    const float* __restrict__ img,
    const float* __restrict__ phi,
    float* __restrict__ out)
{
    // [chan][wave][128 floats] = 12 KB staging for this block's grid slice
    __shared__ float stage[3 * 8 * 128];

    const int tid  = threadIdx.x;
    const int wave = tid >> 5;
    const int lane = tid & 31;
    const int i0   = (blockIdx.x * 256 + tid) * 4;  // 4 voxels along W per lane
    if (i0 + 3 >= N_) return;                        // exact division; never taken

    // ---- async-stage this wave's grid span (128 floats/channel) into LDS ----
    // one global_load_async_to_lds_b128 per channel per wave (32 lanes x 16B)
    {
        const int waveBase = blockIdx.x * 1024 + wave * 128 + lane * 4; // == i0
        const float* g0p = phi + waveBase;
        const float* g1p = g0p + N_;
        const float* g2p = g0p + 2 * N_;
        unsigned l0 = (unsigned)(unsigned long long)&stage[(0 * 8 + wave) * 128 + lane * 4];
        unsigned l1 = (unsigned)(unsigned long long)&stage[(1 * 8 + wave) * 128 + lane * 4];
        unsigned l2 = (unsigned)(unsigned long long)&stage[(2 * 8 + wave) * 128 + lane * 4];
        asm volatile("global_load_async_to_lds_b128 %0, %1, off th:TH_LOAD_NT"
                     :: "v"(l0), "v"(g0p) : "memory");
        asm volatile("global_load_async_to_lds_b128 %0, %1, off th:TH_LOAD_NT"
                     :: "v"(l1), "v"(g1p) : "memory");
        asm volatile("global_load_async_to_lds_b128 %0, %1, off th:TH_LOAD_NT"
                     :: "v"(l2), "v"(g2p) : "memory");
    }

    // Stream-ahead prefetch of the grid into GL2 (global_prefetch_b8).
    __builtin_prefetch(phi + i0 + 8192, 0, 0);
    __builtin_prefetch(phi + N_ + i0 + 8192, 0, 0);
    __builtin_prefetch(phi + 2 * N_ + i0 + 8192, 0, 0);

    // Wait for the async LDS writes of *this wave* (per-wave counter; no
    // cross-wave sharing, so no workgroup barrier is needed).
    asm volatile("s_wait_asynccnt 0x0" ::: "memory");

    f32x4 g0 = *(const f32x4*)&stage[(0 * 8 + wave) * 128 + lane * 4]; // ds_load_b128
    f32x4 g1 = *(const f32x4*)&stage[(1 * 8 + wave) * 128 + lane * 4];
    f32x4 g2 = *(const f32x4*)&stage[(2 * 8 + wave) * 128 + lane * 4];

    // K_d = (2/((sz_d-1)*spacing_d)) * 0.5 * (sz_d-1); spacing = 1/(sz-1) so
    // each K_d folds to (sz_d-1). Faithful expression, constant-folded.
    const float K0 = (2.0f / ((float)(D_ - 1) * (1.0f / 159.0f))) * 0.5f * (float)(D_ - 1);
    const float K1 = (2.0f / ((float)(H_ - 1) * (1.0f / 191.0f))) * 0.5f * (float)(H_ - 1);
    const float K2 = (2.0f / ((float)(W_ - 1) * (1.0f / 223.0f))) * 0.5f * (float)(W_ - 1);

    f32x4 res;
#pragma unroll
    for (int k = 0; k < 4; ++k) {
        float cx = fminf(fmaxf(g0[k] * K0, 0.0f), (float)(D_ - 1));
        float cy = fminf(fmaxf(g1[k] * K1, 0.0f), (float)(H_ - 1));
        float cz = fminf(fmaxf(g2[k] * K2, 0.0f), (float)(W_ - 1));

        // coords are >= 0: integer truncation == floor (no v_floor needed)
        int x0 = (int)cx, y0 = (int)cy, z0 = (int)cz;
        float fx = cx - (float)x0;
        float fy = cy - (float)y0;
        float fz = cz - (float)z0;
        int x1 = (x0 + 1 < D_) ? x0 + 1 : D_ - 1;   // border clamp
        int y1 = (y0 + 1 < H_) ? y0 + 1 : H_ - 1;

        // z-pair trick: shift pair left at the border and fold into the lerp
        // parameter. At z0==W-1, fz==0 (cz clamped to W-1 exactly), so tz==1
        // and the endpoint-exact lerp form returns the right corner exactly.
        int zp = (z0 < W_ - 1) ? z0 : W_ - 2;
        float tz  = fz + (float)(z0 - zp);
        float omz = 1.0f - tz;

        int bx0 = x0 * HW_;
        int bx1 = x1 * HW_;
        int by0 = y0 * W_;
        int by1 = y1 * W_;

        // 4 b64 pair-gathers instead of 8 b32 gathers (L2-resident hits).
        f32x2 a00 = load_pair(img + (bx0 + by0 + zp));
        f32x2 a01 = load_pair(img + (bx0 + by1 + zp));
        f32x2 a10 = load_pair(img + (bx1 + by0 + zp));
        f32x2 a11 = load_pair(img + (bx1 + by1 + zp));

        // z-lerp (endpoint-exact), then y, then x.
        float c00 = fmaf(a00.y, tz, a00.x * omz);
        float c01 = fmaf(a01.y, tz, a01.x * omz);
        float c10 = fmaf(a10.y, tz, a10.x * omz);
        float c11 = fmaf(a11.y, tz, a11.x * omz);
        float c0  = fmaf(fy, c01 - c00, c00);
        float c1  = fmaf(fy, c11 - c10, c10);
        res[k]    = fmaf(fx, c1 - c0, c0);
    }

    // Output written once: non-temporal b128 store preserves L2 for the image.
    __builtin_nontemporal_store(res, (f32x4*)(out + i0));
}

extern "C" void kernel_launch(void* const* d_in, const int* in_sizes, int n_in,
                              void* d_out, int out_size, void* d_ws, size_t ws_size,
                              hipStream_t stream) {
    const float* img = (const float*)d_in[0];   // input1: [1,1,160,192,224]
    const float* phi = (const float*)d_in[1];   // input2: [1,3,160,192,224]
    float* out = (float*)d_out;

    const int threads = 256;                    // 8 wave32 waves per block
    const int quads = N_ / 4;                   // N divisible by 4 (and by 1024)
    const int blocks = (quads + threads - 1) / threads;
    voxelmorph_warp3d_kernel<<<blocks, threads, 0, stream>>>(img, phi, out);
    (void)in_sizes; (void)n_in; (void)out_size; (void)d_ws; (void)ws_size;
}